// SimpleHGN_MTL_Layer_18580028522786
// MI455X (gfx1250) — compile-verified
//
#include <hip/hip_runtime.h>
#include <hip/hip_bf16.h>
#include <math.h>

// ---- problem constants (match reference) ----
#define IN_F     256
#define OUTF     32
#define NHEADS   4
#define PROJ     (OUTF * NHEADS)   // 128
#define EDGE_F   64
#define N_ETYPES 5
#define NEG_SLOPE 0.2f

typedef __attribute__((ext_vector_type(2))) float v2f;
typedef __attribute__((ext_vector_type(8))) float v8f;

#if defined(__has_builtin)
#if __has_builtin(__builtin_amdgcn_wmma_f32_16x16x4_f32)
#define HAVE_WMMA_F32X4 1
#endif
#endif

// monotone float->int key for atomicMax-based segment max
__device__ __forceinline__ int fkey(float f) {
    int i = __float_as_int(f);
    return (i >= 0) ? i : (i ^ 0x7fffffff);
}
__device__ __forceinline__ float fdecode(int k) {
    return __int_as_float((k >= 0) ? k : (k ^ 0x7fffffff));
}
__device__ __forceinline__ float leaky(float x) {
    return x > 0.f ? x : NEG_SLOPE * x;
}

// ---------------------------------------------------------------------------
// K0: zero out-region, init segment-max keys to key(-inf), denom to 0
__global__ void init_kernel(float* __restrict__ out_zero, int n_out,
                            int* __restrict__ mkey, float* __restrict__ denom, int n_seg) {
    int i = blockIdx.x * blockDim.x + threadIdx.x;
    if (i < n_out) out_zero[i] = 0.f;
    if (i < n_seg) { mkey[i] = (int)0x807fffff; denom[i] = 0.f; }
}

// ---------------------------------------------------------------------------
// K1: H = X @ W  ([n,256] x [256,128]) for head (blockIdx.y==0) and tail (==1).
// Block = 256 threads = 8 waves; block computes a 16-row x 128-col stripe,
// wave w computes the 16x16 tile at columns [16w,16w+16) via 64 chained
// v_wmma_f32_16x16x4_f32 ops. A 16x256 tile is staged once in LDS; W stays
// L2-resident (128 KB) so B fragments come straight from global.
__global__ __launch_bounds__(256) void gemm_proj_kernel(
        const float* __restrict__ Xhead, const float* __restrict__ Xtail,
        const float* __restrict__ W,
        float* __restrict__ Hhead, float* __restrict__ Htail, int nrows) {
    const float* X = blockIdx.y ? Xtail : Xhead;
    float*       H = blockIdx.y ? Htail : Hhead;

    const int row0 = blockIdx.x * 16;
    const int tid  = threadIdx.x;
    const int lane = tid & 31;
    const int wave = tid >> 5;
    const int col0 = wave * 16;

    __shared__ float As[16 * 260];   // padded row stride to spread LDS banks

    // cooperative load of A tile (16 rows x 256 f32), float4 coalesced
    for (int i = tid; i < 16 * 64; i += 256) {
        int r  = i >> 6;
        int c4 = i & 63;
        int rr = row0 + r; if (rr >= nrows) rr = nrows - 1;   // clamp (n%16==0 normally)
        float4 v = *(const float4*)(X + (size_t)rr * IN_F + c4 * 4);
        *(float4*)(&As[r * 260 + c4 * 4]) = v;
    }
    __syncthreads();

    const int l15   = lane & 15;
    const int khalf = (lane < 16) ? 0 : 2;   // K sub-offset for A/B fragments
    const int mrow  = (lane < 16) ? 0 : 8;   // C/D row base

    v8f c = {0.f, 0.f, 0.f, 0.f, 0.f, 0.f, 0.f, 0.f};

#ifdef HAVE_WMMA_F32X4
    for (int k0 = 0; k0 < IN_F; k0 += 4) {
        v2f a, b;
        // A 16x4: lane<16 -> (M=l15, K=k0+{0,1}) ; lane>=16 -> K=k0+{2,3}
        a.x = As[l15 * 260 + k0 + khalf];
        a.y = As[l15 * 260 + k0 + khalf + 1];
        // B 4x16: lane<16 -> (K=k0+{0,1}, N=l15) ; lane>=16 -> K=k0+{2,3}
        const float* Wp = W + (size_t)(k0 + khalf) * PROJ + col0 + l15;
        b.x = Wp[0];
        b.y = Wp[PROJ];
        c = __builtin_amdgcn_wmma_f32_16x16x4_f32(false, a, false, b,
                                                  (short)0, c, false, false);
    }
#else
    // scalar fallback with identical C layout
    for (int k = 0; k < IN_F; ++k) {
        float bx = W[(size_t)k * PROJ + col0 + l15];
        #pragma unroll
        for (int r = 0; r < 8; ++r)
            c[r] = fmaf(As[(mrow + r) * 260 + k], bx, c[r]);
    }
#endif

    // C/D layout: VGPR r -> (M = mrow + r, N = l15)
    float* Hp = H + ((size_t)(row0 + mrow)) * PROJ + col0 + l15;
    if (row0 + 16 <= nrows) {
        // full tile (always taken when nrows % 16 == 0): branchless stores
        #pragma unroll
        for (int r = 0; r < 8; ++r)
            Hp[(size_t)r * PROJ] = c[r];
    } else {
        #pragma unroll
        for (int r = 0; r < 8; ++r)
            if (row0 + mrow + r < nrows)
                Hp[(size_t)r * PROJ] = c[r];
    }
}

// ---------------------------------------------------------------------------
// K2: hl[i,h] = sum_d a_l[h,d] * Hhead[i, h*32+d]   (and hr from tail)
__global__ void hlr_kernel(const float* __restrict__ Hh, const float* __restrict__ Ht,
                           const float* __restrict__ a_l, const float* __restrict__ a_r,
                           float* __restrict__ hl, float* __restrict__ hr, int n) {
    int idx = blockIdx.x * blockDim.x + threadIdx.x;
    int per = n * NHEADS;
    if (idx >= 2 * per) return;
    int which = idx >= per;
    int r = idx - which * per;
    int i = r >> 2, h = r & 3;
    const float* H = which ? Ht : Hh;
    const float* a = which ? a_r : a_l;
    const float* hp = H + (size_t)i * PROJ + h * OUTF;
    const float* ap = a + h * OUTF;
    float s = 0.f;
    #pragma unroll
    for (int d = 0; d < OUTF; d += 4) {
        float4 hv = *(const float4*)(hp + d);
        float4 av = *(const float4*)(ap + d);
        s += hv.x * av.x + hv.y * av.y + hv.z * av.z + hv.w * av.w;
    }
    (which ? hr : hl)[r] = s;
}

// ---------------------------------------------------------------------------
// K3: he[t,h] = sum_d a_e[h,d] * (edge_emb @ W_e)[t, h*64+d]   (tiny)
__global__ void he_kernel(const float* __restrict__ edge_emb, const float* __restrict__ W_e,
                          const float* __restrict__ a_e, float* __restrict__ he) {
    int idx = threadIdx.x;
    if (idx >= N_ETYPES * NHEADS) return;
    int t = idx >> 2, h = idx & 3;
    float s = 0.f;
    for (int d = 0; d < EDGE_F; ++d) {
        int cc = h * EDGE_F + d;
        float ec = 0.f;
        for (int k = 0; k < EDGE_F; ++k)
            ec = fmaf(edge_emb[t * EDGE_F + k], W_e[(size_t)k * (EDGE_F * NHEADS) + cc], ec);
        s = fmaf(a_e[h * EDGE_F + d], ec, s);
    }
    he[idx] = s;
}

// ---------------------------------------------------------------------------
// K4: att[e,h] = leaky(hl[head]+hr[tail]+he[type]); store + segment atomicMax
__global__ void att_kernel(const int* __restrict__ head_ind, const int* __restrict__ tail_ind,
                           const int* __restrict__ etype,
                           const float* __restrict__ hl, const float* __restrict__ hr,
                           const float* __restrict__ he,
                           float* __restrict__ att_out, int* __restrict__ mkey, int E) {
    int e = blockIdx.x * blockDim.x + threadIdx.x;
    if (e >= E) return;
    int hi = head_ind[e], ti = tail_ind[e], ty = etype[e];
    float4 l = *(const float4*)(hl + (size_t)hi * 4);
    float4 r = *(const float4*)(hr + (size_t)ti * 4);
    float4 g = *(const float4*)(he + (size_t)ty * 4);
    float4 a;
    a.x = leaky(l.x + r.x + g.x);
    a.y = leaky(l.y + r.y + g.y);
    a.z = leaky(l.z + r.z + g.z);
    a.w = leaky(l.w + r.w + g.w);
    *(float4*)(att_out + (size_t)e * 4) = a;
    int* mp = mkey + (size_t)hi * 4;
    atomicMax(mp + 0, fkey(a.x));
    atomicMax(mp + 1, fkey(a.y));
    atomicMax(mp + 2, fkey(a.z));
    atomicMax(mp + 3, fkey(a.w));
}

// ---------------------------------------------------------------------------
// K5: ex = exp(att - m[head]); in-place; denom[head] += ex (f32 atomics in L2)
__global__ void ex_kernel(const int* __restrict__ head_ind, const int* __restrict__ mkey,
                          float* __restrict__ att_io, float* __restrict__ denom, int E) {
    int e = blockIdx.x * blockDim.x + threadIdx.x;
    if (e >= E) return;
    int hi = head_ind[e];
    float4 a = *(const float4*)(att_io + (size_t)e * 4);
    int4  mk = *(const int4*)(mkey + (size_t)hi * 4);
    float4 x;
    x.x = expf(a.x - fdecode(mk.x));
    x.y = expf(a.y - fdecode(mk.y));
    x.z = expf(a.z - fdecode(mk.z));
    x.w = expf(a.w - fdecode(mk.w));
    *(float4*)(att_io + (size_t)e * 4) = x;
    float* dp = denom + (size_t)hi * 4;
    atomicAdd(dp + 0, x.x);
    atomicAdd(dp + 1, x.y);
    atomicAdd(dp + 2, x.z);
    atomicAdd(dp + 3, x.w);
}

// ---------------------------------------------------------------------------
// K6: one wave32 per edge: alpha = ex/denom (written to alpha output) and
//     out[head, :] += alpha[h] * h_tail[tail, :]   (128 f32 atomics per edge)
__global__ __launch_bounds__(256) void agg_kernel(
        const int* __restrict__ head_ind, const int* __restrict__ tail_ind,
        const float* __restrict__ denom, const float* __restrict__ Ht,
        float* __restrict__ alpha_io, float* __restrict__ out, int E) {
    int gid  = blockIdx.x * blockDim.x + threadIdx.x;
    int e    = gid >> 5;
    int lane = threadIdx.x & 31;
    if (e >= E) return;
    int hi = head_ind[e], ti = tail_ind[e];
    float4 ex4 = *(const float4*)(alpha_io + (size_t)e * 4);
    float4 dn  = *(const float4*)(denom + (size_t)hi * 4);
    float4 al;
    al.x = ex4.x / dn.x;  al.y = ex4.y / dn.y;
    al.z = ex4.z / dn.z;  al.w = ex4.w / dn.w;
    if (lane < 4) {
        float v = (lane == 0) ? al.x : (lane == 1) ? al.y : (lane == 2) ? al.z : al.w;
        alpha_io[(size_t)e * 4 + lane] = v;
    }
    const float* ht = Ht  + (size_t)ti * PROJ;
    float*       op = out + (size_t)hi * PROJ;
    atomicAdd(op + lane,      al.x * ht[lane]);
    atomicAdd(op + 32 + lane, al.y * ht[32 + lane]);
    atomicAdd(op + 64 + lane, al.z * ht[64 + lane]);
    atomicAdd(op + 96 + lane, al.w * ht[96 + lane]);
}

// ---------------------------------------------------------------------------
// K7: out = elu(out) in place
__global__ void elu_kernel(float* __restrict__ out, int n) {
    int i = blockIdx.x * blockDim.x + threadIdx.x;
    if (i >= n) return;
    float x = out[i];
    out[i] = (x > 0.f) ? x : expm1f(x);
}

// ---------------------------------------------------------------------------
extern "C" void kernel_launch(void* const* d_in, const int* in_sizes, int n_in,
                              void* d_out, int out_size, void* d_ws, size_t ws_size,
                              hipStream_t stream) {
    const float* head_feature = (const float*)d_in[0];
    const float* tail_feature = (const float*)d_in[1];
    const float* W            = (const float*)d_in[2];
    const float* W_e          = (const float*)d_in[3];
    const float* edge_emb     = (const float*)d_in[4];
    const float* a_l          = (const float*)d_in[5];
    const float* a_r          = (const float*)d_in[6];
    const float* a_e          = (const float*)d_in[7];
    const int*   edge_list    = (const int*)d_in[8];
    const int*   tmp_edge     = (const int*)d_in[9];

    const int N = in_sizes[0] / IN_F;       // 50000
    const int E = in_sizes[9];              // 1600000
    const int* head_ind = edge_list;
    const int* tail_ind = edge_list + E;

    float* out   = (float*)d_out;                   // [N, 128]
    float* alpha = out + (size_t)N * PROJ;          // [E, 4]

    // workspace layout (floats)
    float* ws     = (float*)d_ws;
    float* h_head = ws;                                   // N*128
    float* h_tail = h_head + (size_t)N * PROJ;            // N*128
    float* hl     = h_tail + (size_t)N * PROJ;            // N*4
    float* hr     = hl + (size_t)N * NHEADS;              // N*4
    float* he     = hr + (size_t)N * NHEADS;              // 32 (padded)
    int*   mkey   = (int*)(he + 32);                      // N*4
    float* denom  = (float*)(mkey + (size_t)N * NHEADS);  // N*4

    const int n_out = N * PROJ;
    const int n_seg = N * NHEADS;

    init_kernel<<<(n_out + 255) / 256, 256, 0, stream>>>(out, n_out, mkey, denom, n_seg);

    gemm_proj_kernel<<<dim3((N + 15) / 16, 2), 256, 0, stream>>>(
        head_feature, tail_feature, W, h_head, h_tail, N);

    hlr_kernel<<<(2 * n_seg + 255) / 256, 256, 0, stream>>>(
        h_head, h_tail, a_l, a_r, hl, hr, N);

    he_kernel<<<1, 32, 0, stream>>>(edge_emb, W_e, a_e, he);

    att_kernel<<<(E + 255) / 256, 256, 0, stream>>>(
        head_ind, tail_ind, tmp_edge, hl, hr, he, alpha, mkey, E);

    ex_kernel<<<(E + 255) / 256, 256, 0, stream>>>(head_ind, mkey, alpha, denom, E);

    long long aggThreads = (long long)E * 32;
    agg_kernel<<<(unsigned)((aggThreads + 255) / 256), 256, 0, stream>>>(
        head_ind, tail_ind, denom, h_tail, alpha, out, E);

    elu_kernel<<<(n_out + 255) / 256, 256, 0, stream>>>(out, n_out);
}